// EventWarping_83210696393076
// MI455X (gfx1250) — compile-verified
//
#include <hip/hip_runtime.h>
#include <math.h>

// Problem constants (match reference setup_inputs)
#define RES_H   256
#define RES_W   256
#define NPIX    (RES_H * RES_W)      // 65536
#define NBATCH  8
#define NEV     262144
#define EPSF    1e-9f
#define FTR     1e-3f                 // FLOW_TEMP_REG
#define NGROUP  16                    // 2 warps * 8 batches

// Workspace layout (floats):
//   [0, ACC_FLTS)   : accumulators, float4 per (group, pixel):
//                     .x = pos_w  .y = pos_w*tsw  .z = neg_w  .w = neg_w*tsw
//   [ACC_FLTS, +32) : per-group partials {sumsq, nzcount} x 16 groups
static constexpr unsigned long long ACC_F4   = 2ull * NBATCH * NPIX;  // 1,048,576 float4
static constexpr unsigned long long ACC_FLTS = ACC_F4 * 4;            // 4,194,304 floats (16 MB)

__device__ __forceinline__ void ew_splat(float* __restrict__ base, int cy, int cx,
                                         float w, float tsw, int cbase) {
    // Skipping w==0 / out-of-bounds corners is exact: reference adds +0.0 for them.
    if (cy >= 0 && cy < RES_H && cx >= 0 && cx < RES_W && w > 0.f) {
        float* p = base + (((unsigned)(cy * RES_W + cx)) << 2) + cbase;
        // Non-returning fp32 global atomics -> L2-side global_atomic_add_f32.
        __hip_atomic_fetch_add(p,     w,       __ATOMIC_RELAXED, __HIP_MEMORY_SCOPE_AGENT);
        __hip_atomic_fetch_add(p + 1, w * tsw, __ATOMIC_RELAXED, __HIP_MEMORY_SCOPE_AGENT);
    }
}

// Hot kernel — kept first in the file so the disasm snippet shows it.
__global__ void ew_scatter(const float4* __restrict__ ev,   // [B*N] (ts,y,x,p)
                           const float2* __restrict__ fl,   // [B*N] (fy,fx)
                           const int*    __restrict__ mts,
                           float*        __restrict__ acc) {
    unsigned gid = blockIdx.x * blockDim.x + threadIdx.x;
    if (gid >= (unsigned)(NBATCH * NEV)) return;
    const unsigned b = gid >> 18;                 // NEV == 2^18
    const float4 e = ev[gid];                     // global_load_b128
    const float2 f = fl[gid];                     // global_load_b64
    const float  mt = (float)mts[0];
    const float  ts = e.x, y0 = e.y, x0 = e.z, p = e.w;
    const int cbase = (p > 0.f) ? 0 : 2;          // pol_mask is one-hot on sign(p)

#pragma unroll
    for (int wrp = 0; wrp < 2; ++wrp) {
        const float tref = (wrp == 0) ? mt : 0.f;
        const float tsw  = (wrp == 0) ? ts : (mt - ts);
        const float dt   = tref - ts;
        const float wy = y0 + dt * f.x;
        const float wx = x0 + dt * f.y;
        const float fy = floorf(wy), fx = floorf(wx);
        const float ay = wy - fy,    ax = wx - fx;     // frac in [0,1)
        const int iy = (int)fy, ix = (int)fx;
        float* base = acc + ((unsigned long long)(wrp * NBATCH + b) * NPIX) * 4ull;
        const float by = 1.f - ay, bx = 1.f - ax;
        ew_splat(base, iy,     ix,     by * bx, tsw, cbase);  // top-left
        ew_splat(base, iy,     ix + 1, by * ax, tsw, cbase);  // top-right
        ew_splat(base, iy + 1, ix,     ay * bx, tsw, cbase);  // bottom-left
        ew_splat(base, iy + 1, ix + 1, ay * ax, tsw, cbase);  // bottom-right
    }
}

__global__ void ew_zero(float4* __restrict__ acc4, float* __restrict__ part) {
    unsigned long long i = (unsigned long long)blockIdx.x * blockDim.x + threadIdx.x;
    if (i < ACC_F4) acc4[i] = make_float4(0.f, 0.f, 0.f, 0.f);
    if (i < 2 * NGROUP) part[i] = 0.f;
}

__device__ __forceinline__ float ew_wave_red(float v) {
#pragma unroll
    for (int o = 16; o > 0; o >>= 1) v += __shfl_down(v, o, 32);
    return v;
}

// 512 blocks x 256 threads: 32 blocks per (warp,batch) group, 8 pixels/thread.
__global__ void ew_reduce(const float4* __restrict__ acc4,
                          float* __restrict__ part,
                          const int* __restrict__ mts) {
    const int group = blockIdx.x >> 5;
    const int sub   = blockIdx.x & 31;
    const float inv_mt = 1.f / (float)mts[0];
    const unsigned long long base =
        (unsigned long long)group * NPIX + (unsigned long long)sub * 2048ull;

    float sumsq = 0.f, nz = 0.f;
#pragma unroll
    for (int i = 0; i < 8; ++i) {
        const float4 a = acc4[base + (unsigned)(i * 256 + threadIdx.x)];
        const float pt = (a.y / (a.x + EPSF)) * inv_mt;  // iwe_pos_ts
        const float nt = (a.w / (a.z + EPSF)) * inv_mt;  // iwe_neg_ts
        sumsq += pt * pt + nt * nt;
        nz    += ((a.x + a.z) != 0.f) ? 1.f : 0.f;
    }

    __shared__ float ss[8], sn[8];
    const int lane = threadIdx.x & 31, wid = threadIdx.x >> 5;
    sumsq = ew_wave_red(sumsq);
    nz    = ew_wave_red(nz);
    if (lane == 0) { ss[wid] = sumsq; sn[wid] = nz; }
    __syncthreads();
    if (wid == 0) {
        float s = (lane < 8) ? ss[lane] : 0.f;
        float n = (lane < 8) ? sn[lane] : 0.f;
#pragma unroll
        for (int o = 4; o > 0; o >>= 1) { s += __shfl_down(s, o, 32); n += __shfl_down(n, o, 32); }
        if (lane == 0) {
            __hip_atomic_fetch_add(&part[group * 2],     s, __ATOMIC_RELAXED, __HIP_MEMORY_SCOPE_AGENT);
            __hip_atomic_fetch_add(&part[group * 2 + 1], n, __ATOMIC_RELAXED, __HIP_MEMORY_SCOPE_AGENT);
        }
    }
}

// One wave32: per-group loss + Charbonnier temporal smoothness, write scalar.
__global__ void ew_finalize(const float* __restrict__ part,
                            const float* __restrict__ vec,  // [B,4,8]
                            float* __restrict__ out) {
    const int l = threadIdx.x;
    float v = 0.f;
    if (l < NGROUP) {
        const float s = part[l * 2], n = part[l * 2 + 1];
        v = s / (n + EPSF);                      // LOSS_SCALING
    }
    float c = 0.f;
    if (l < NBATCH) {                            // lane b: Charbonnier mean for batch b
        const float* vb = vec + l * 4 * 8;
        float sum = 0.f;
#pragma unroll
        for (int i = 0; i < 3; ++i)
#pragma unroll
            for (int j = 0; j < 8; ++j) {
                const float d = vb[i * 8 + j] - vb[(i + 1) * 8 + j];
                sum += sqrtf(d * d + EPSF);
            }
        c = FTR * (sum * (1.f / 24.f));
    }
    float t = v + c;
#pragma unroll
    for (int o = 16; o > 0; o >>= 1) t += __shfl_down(t, o, 32);
    if (l == 0) out[0] = t;
}

extern "C" void kernel_launch(void* const* d_in, const int* in_sizes, int n_in,
                              void* d_out, int out_size, void* d_ws, size_t ws_size,
                              hipStream_t stream) {
    (void)in_sizes; (void)n_in; (void)out_size; (void)ws_size;
    const float4* ev  = (const float4*)d_in[0];  // event_list [B,N,4] f32
    const float2* fl  = (const float2*)d_in[1];  // flow       [B,N,2] f32
    // d_in[2] = pol_mask: derivable exactly from sign(p); not needed.
    const float*  vec = (const float*)d_in[3];   // vector_list [B,4,8] f32
    const int*    mts = (const int*)d_in[4];     // max_ts (python int)
    float* ws    = (float*)d_ws;
    float4* acc4 = (float4*)ws;
    float* part  = ws + ACC_FLTS;
    float* out   = (float*)d_out;

    ew_zero<<<(unsigned)((ACC_F4 + 255) / 256), 256, 0, stream>>>(acc4, part);
    ew_scatter<<<(NBATCH * NEV) / 256, 256, 0, stream>>>(ev, fl, mts, ws);
    ew_reduce<<<NGROUP * 32, 256, 0, stream>>>(acc4, part, mts);
    ew_finalize<<<1, 32, 0, stream>>>(part, vec, out);
}